// LinkPredictor_55929064129416
// MI455X (gfx1250) — compile-verified
//
#include <hip/hip_runtime.h>

typedef __attribute__((ext_vector_type(2))) float v2f;
typedef __attribute__((ext_vector_type(8))) float v8f;

#define N_ENT  100000
#define DIMK   128
#define BQ     256      // batch
#define NF     20000
#define EPSF   1e-12f

#define ENT_TILE   128
#define BLOCK      256
#define LDS_STRIDE 132  // floats per entity row in LDS: 8B-aligned, conflict-free

// workspace layout (float elements)
#define WS_QW 0
#define WS_PW (BQ * DIMK)          // 32768
#define WS_QQ (2 * BQ * DIMK)      // 65536
#define WS_PP (WS_QQ + BQ)
#define WS_TC (WS_PP + BQ)

// low 32 bits of a generic pointer to LDS == LDS byte offset (aperture in hi32)
__device__ __forceinline__ unsigned lds_off(const void* p) {
  return (unsigned)(unsigned long long)p;
}

__device__ __forceinline__ void async_ld_b128(unsigned lds, unsigned goff,
                                              const void* sbase) {
  asm volatile("global_load_async_to_lds_b128 %0, %1, %2"
               :: "v"(lds), "v"(goff), "s"(sbase) : "memory");
}

__device__ __forceinline__ void wait_async0() {
  asm volatile("s_wait_asynccnt 0" ::: "memory");
}

// ---------------------------------------------------------------------------
// prep: build Q/P in WMMA A-fragment order, norms, thresholds, init ranks = 1
// A-fragment layout (16x4 f32): element [m][k] lives at lane = (m&15)+16*(k>>1),
// vgpr = k&1.  Flat float2 slot for (row-tile mt, chunk j): (mt*32+j)*32 + lane.
// ---------------------------------------------------------------------------
__global__ void __launch_bounds__(BQ) prep_kernel(
    const float* __restrict__ ent, const float* __restrict__ rel,
    const int* __restrict__ heads, const int* __restrict__ rels,
    const int* __restrict__ tails, float* __restrict__ ws,
    float* __restrict__ out) {
  int b = threadIdx.x;
  if (b >= BQ) return;
  const float* eh = ent + (long)heads[b] * DIMK;
  const float* er = rel + (long)rels[b]  * DIMK;
  const float* et = ent + (long)tails[b] * DIMK;
  float qq = 0.f, pp = 0.f, T = 0.f;
  float* Qw = ws + WS_QW;
  float* Pw = ws + WS_PW;
  const int mt = b >> 4, mlo = b & 15;
  for (int d = 0; d < DIMK; ++d) {
    float q = er[d] - et[d];
    float p = eh[d] + er[d];
    float df = p - et[d];
    qq += q * q; pp += p * p; T += df * df;
    int j = d >> 2, k2 = d & 3;
    int lane = mlo + 16 * (k2 >> 1);
    int off = (((mt * 32 + j) * 32) + lane) * 2 + (k2 & 1);
    Qw[off] = q; Pw[off] = p;
  }
  ws[WS_QQ + b] = qq;
  ws[WS_PP + b] = pp;
  ws[WS_TC + b] = fmaxf(T, EPSF);
  out[b]      = 1.0f;   // rank = count + 1
  out[BQ + b] = 1.0f;
}

// ---------------------------------------------------------------------------
// main: block = 128-entity tile; async-staged fp32 WMMA GEMM + fused ranking
// ---------------------------------------------------------------------------
__global__ void __launch_bounds__(BLOCK) rank_kernel(
    const float* __restrict__ ent,
    const int* __restrict__ heads, const int* __restrict__ rels,
    const int* __restrict__ tails,
    const int* __restrict__ fh, const int* __restrict__ fr,
    const int* __restrict__ ft,
    const float* __restrict__ ws, float* __restrict__ out) {
  __shared__ __align__(16) float tile[ENT_TILE * LDS_STRIDE]; // 67584 B
  __shared__ __align__(16) float astage[2][4096];             // 32768 B (Q|P per buf)
  __shared__ float    sq[ENT_TILE];
  __shared__ float    qqs[BQ], pps[BQ], tcs[BQ];
  __shared__ int      hs[BQ], rs[BQ], ts[BQ];
  __shared__ unsigned maskH[BQ * 4], maskT[BQ * 4];           // 128 bits/row/dir
  __shared__ unsigned cntH[BQ], cntT[BQ];

  const int t    = threadIdx.x;
  const int base = blockIdx.x * ENT_TILE;
  const float* Qglob = ws + WS_QW;
  const float* Pglob = ws + WS_PW;

  // ---- phase A: init LDS metadata, kick off async fills, build masks ----
  for (int b = t; b < BQ; b += BLOCK) {
    qqs[b] = ws[WS_QQ + b]; pps[b] = ws[WS_PP + b]; tcs[b] = ws[WS_TC + b];
    hs[b] = heads[b]; rs[b] = rels[b]; ts[b] = tails[b];
    cntH[b] = 0u; cntT[b] = 0u;
#pragma unroll
    for (int k = 0; k < 4; ++k) { maskH[4 * b + k] = 0u; maskT[4 * b + k] = 0u; }
  }
  if (t < ENT_TILE) sq[t] = 0.f;

  // zero-fill rows past N_ENT (only last block; async loads skip those lanes)
  {
    int oob = N_ENT - base;   // first OOB local row
    if (oob < ENT_TILE) {
      for (int nl = (oob < 0 ? 0 : oob); nl < ENT_TILE; ++nl)
        for (int d = t; d < LDS_STRIDE; d += BLOCK) tile[nl * LDS_STRIDE + d] = 0.f;
    }
  }

  // async: entity tile -> LDS (coalesced b128, padded rows)
  const unsigned tile_lds = lds_off(tile);
#pragma unroll
  for (int rep = 0; rep < 16; ++rep) {
    int u  = t + BLOCK * rep;            // 4096 float4 slots
    int nl = u >> 5, d4 = u & 31;
    int ng = base + nl;
    if (ng < N_ENT)
      async_ld_b128(tile_lds + (unsigned)(nl * LDS_STRIDE + d4 * 4) * 4u,
                    (unsigned)ng * (DIMK * 4u) + (unsigned)d4 * 16u, ent);
  }
  // async: stage A fragments for mt = 0 into buffer 0 (Q at 0, P at +8KB)
  {
    const unsigned s_lds = lds_off(&astage[0][0]);
#pragma unroll
    for (int it = 0; it < 2; ++it) {
      unsigned go = (unsigned)it * 4096u + (unsigned)t * 16u;
      async_ld_b128(s_lds + go,         go, Qglob);
      async_ld_b128(s_lds + 8192u + go, go, Pglob);
    }
  }

  // filter masks (overlaps with async fills)
  for (int f = t; f < NF; f += BLOCK) {
    int a = fh[f], r = fr[f], c = ft[f];
    int al = a - base, cl = c - base;
    bool ain = (unsigned)al < (unsigned)ENT_TILE;
    bool cin = (unsigned)cl < (unsigned)ENT_TILE;
    if (ain | cin) {
      for (int b = 0; b < BQ; ++b) {
        if (rs[b] == r) {
          if (ain && ts[b] == c) atomicOr(&maskH[4 * b + (al >> 5)], 1u << (al & 31));
          if (cin && hs[b] == a) atomicOr(&maskT[4 * b + (cl >> 5)], 1u << (cl & 31));
        }
      }
    }
  }
  wait_async0();
  __syncthreads();

  // ---- phase B: per-tile ||e_n||^2 ----
  {
    int e = t & (ENT_TILE - 1), half = t >> 7;
    const float* row = &tile[e * LDS_STRIDE + half * 64];
    float s = 0.f;
    for (int d = 0; d < 64; ++d) s += row[d] * row[d];
    atomicAdd(&sq[e], s);
  }
  __syncthreads();

  // ---- phase C: WMMA GEMM + fused ranking, double-buffered A staging ----
  const int  lane   = t & 31;
  const int  w      = t >> 5;            // wave id: entities [w*16, w*16+16)
  const int  nlo    = lane & 15;
  const int  h      = lane >> 4;
  const int  nlocal = w * 16 + nlo;
  const int  ng     = base + nlocal;
  const float entsq = sq[nlocal];
  const bool inrange = (ng < N_ENT);
  const unsigned bit = 1u << (nlocal & 31);
  const int  word    = nlocal >> 5;

  // B-fragment (4x16): element [k][n] at lane = n + 16*(k>>1), vgpr = k&1.
  // Preload all 32 chunks into registers: they are mt-invariant, and the
  // async-asm "memory" clobber would otherwise force per-mt LDS reloads.
  const float* bbase = &tile[nlocal * LDS_STRIDE + 2 * h];
  v2f bfr[32];
#pragma unroll
  for (int j = 0; j < 32; ++j) bfr[j] = *(const v2f*)(bbase + 4 * j);

  for (int mt = 0; mt < 16; ++mt) {      // 16 row-tiles cover B = 256
    if (mt) { wait_async0(); __syncthreads(); }
    if (mt < 15) {                       // prefetch A chunk for mt+1
      const unsigned s_lds = lds_off(&astage[(mt + 1) & 1][0]);
      const float* qsrc = Qglob + (size_t)(mt + 1) * 2048;
      const float* psrc = Pglob + (size_t)(mt + 1) * 2048;
#pragma unroll
      for (int it = 0; it < 2; ++it) {
        unsigned go = (unsigned)it * 4096u + (unsigned)t * 16u;
        async_ld_b128(s_lds + go,         go, qsrc);
        async_ld_b128(s_lds + 8192u + go, go, psrc);
      }
    }

    const v2f* aqp = (const v2f*)&astage[mt & 1][0];     // Q frags
    const v2f* app = aqp + 1024;                         // P frags (+8KB)
    v8f accq = {0.f, 0.f, 0.f, 0.f, 0.f, 0.f, 0.f, 0.f};
    v8f accp = accq;
#pragma unroll
    for (int j = 0; j < 32; ++j) {       // K = 128 in chunks of 4
      v2f aq = aqp[j * 32 + lane];
      v2f ap = app[j * 32 + lane];
      accq = __builtin_amdgcn_wmma_f32_16x16x4_f32(false, aq, false, bfr[j],
                                                   (short)0, accq, false, false);
      accp = __builtin_amdgcn_wmma_f32_16x16x4_f32(false, ap, false, bfr[j],
                                                   (short)0, accp, false, false);
    }
#pragma unroll
    for (int i = 0; i < 8; ++i) {
      // C/D layout: vgpr i, lanes 0-15 -> M=i ; lanes 16-31 -> M=i+8
      int m = mt * 16 + i + 8 * h;
      float qqm = qqs[m], ppm = pps[m], tcm = tcs[m];
      unsigned hm = maskH[4 * m + word], tm = maskT[4 * m + word];
      float aH = entsq + 2.f * accq[i] + qqm;   // head corruption sq-dist
      float aT = entsq - 2.f * accp[i] + ppm;   // tail corruption sq-dist
      bool cH = inrange && !(hm & bit) && (fmaxf(aH, EPSF) < tcm);
      bool cT = inrange && !(tm & bit) && (fmaxf(aT, EPSF) < tcm);
      unsigned long long bH = __ballot(cH);
      unsigned long long bT = __ballot(cT);
      if (lane == 0) {
        atomicAdd(&cntH[mt * 16 + i],     (unsigned)__popcll(bH & 0xFFFFull));
        atomicAdd(&cntH[mt * 16 + i + 8], (unsigned)__popcll((bH >> 16) & 0xFFFFull));
        atomicAdd(&cntT[mt * 16 + i],     (unsigned)__popcll(bT & 0xFFFFull));
        atomicAdd(&cntT[mt * 16 + i + 8], (unsigned)__popcll((bT >> 16) & 0xFFFFull));
      }
    }
  }
  __syncthreads();
  for (int b = t; b < BQ; b += BLOCK) {
    atomicAdd(&out[b],      (float)cntH[b]);   // exact: integer-valued floats
    atomicAdd(&out[BQ + b], (float)cntT[b]);
  }
}

// ---------------------------------------------------------------------------
extern "C" void kernel_launch(void* const* d_in, const int* in_sizes, int n_in,
                              void* d_out, int out_size, void* d_ws, size_t ws_size,
                              hipStream_t stream) {
  const float* ent   = (const float*)d_in[0];
  const float* rel   = (const float*)d_in[1];
  const int*   heads = (const int*)d_in[2];
  const int*   rels  = (const int*)d_in[3];
  const int*   tails = (const int*)d_in[4];
  const int*   fhp   = (const int*)d_in[5];
  const int*   frp   = (const int*)d_in[6];
  const int*   ftp   = (const int*)d_in[7];
  float* ws  = (float*)d_ws;
  float* out = (float*)d_out;

  prep_kernel<<<1, BQ, 0, stream>>>(ent, rel, heads, rels, tails, ws, out);

  int nblocks = (N_ENT + ENT_TILE - 1) / ENT_TILE;   // 782
  rank_kernel<<<nblocks, BLOCK, 0, stream>>>(ent, heads, rels, tails,
                                             fhp, frp, ftp, ws, out);
}